// HashEncoder_2534030705130
// MI455X (gfx1250) — compile-verified
//
#include <hip/hip_runtime.h>
#include <cmath>
#include <cstdint>
#include <cstddef>

#define NTHREADS 512
#define PPT      8   // points per thread

static constexpr uint32_t kB = 2097152u;   // number of query points

// Per-level resolutions (torch-ngp construction, verified against reference math)
constexpr uint32_t kRes[16] = {16u, 23u, 31u, 43u, 59u, 81u, 112u, 154u,
                               213u, 295u, 407u, 562u, 777u, 1073u, 1483u, 2048u};
// Cumulative embedding offsets (entries). Levels 0-4 dense (8-aligned res^3),
// levels 5-15 hashed with 2^19 entries each. Total = 6,098,120.
constexpr uint32_t kOff[16] = {0u, 4096u, 16264u, 46056u, 125568u, 330952u,
                               855240u, 1379528u, 1903816u, 2428104u, 2952392u,
                               3476680u, 4000968u, 4525256u, 5049544u, 5573832u};
constexpr uint32_t kHashMask = 524287u;    // hashmap = 2^19 for all hashed levels
constexpr uint32_t kLdsElems = 16264u;     // float2 entries for levels 0+1 (130,112 B)

struct ScaleParams { float s[16]; };

typedef float __attribute__((ext_vector_type(4))) v4f;
typedef float __attribute__((ext_vector_type(2))) v2f;
typedef int   __attribute__((ext_vector_type(4))) v4i;
typedef __attribute__((address_space(1))) v4i* g_v4i_p;   // global int4*
typedef __attribute__((address_space(3))) v4i* l_v4i_p;   // LDS int4*

__device__ __forceinline__ uint32_t umin32(uint32_t a, uint32_t b) { return a < b ? a : b; }

__device__ __forceinline__ float fract_dev(float v) {
#if __has_builtin(__builtin_amdgcn_fractf)
  return __builtin_amdgcn_fractf(v);   // v_fract_f32: v - floor(v)
#else
  return v - floorf(v);
#endif
}

template <int LV>
__device__ __forceinline__ void do_level(float x, float y, float z, float scale,
                                         const char* __restrict__ gtab_bytes,
                                         const char* ltab_bytes, float* outv) {
  constexpr uint32_t res    = kRes[LV];
  constexpr bool     hashed = (LV >= 5);
  constexpr bool     useLds = (LV <= 1);
  constexpr uint32_t ldsByteOff = (LV == 0) ? 0u : (4096u * 8u);

  const float px = fmaf(x, scale, 0.5f);
  const float py = fmaf(y, scale, 0.5f);
  const float pz = fmaf(z, scale, 0.5f);
  // pos >= 0.5, so u32 truncation == floor, and v_fract gives the frac part.
  const uint32_t ix = (uint32_t)px, iy = (uint32_t)py, iz = (uint32_t)pz;
  const float tx = fract_dev(px), ty = fract_dev(py), tz = fract_dev(pz);

  const uint32_t rm1 = res - 1u;
  const uint32_t cx0 = umin32(ix, rm1), cx1 = umin32(ix + 1u, rm1);
  const uint32_t cy0 = umin32(iy, rm1), cy1 = umin32(iy + 1u, rm1);
  const uint32_t cz0 = umin32(iz, rm1), cz1 = umin32(iz + 1u, rm1);

  // Byte-offset terms with the *8 (sizeof(float2)) folded into the shifts /
  // multiply constants (shift distributes over xor/add and over the mask).
  const uint32_t hx8_0 = cx0 << 3, hx8_1 = cx1 << 3;
  uint32_t exy8[4], ez8_0, ez8_1;
  if constexpr (hashed) {
    constexpr uint32_t K8y = 2654435761u * 8u;   // mod 2^32
    constexpr uint32_t K8z = 805459861u * 8u;    // mod 2^32
    const uint32_t hy8_0 = cy0 * K8y, hy8_1 = cy1 * K8y;
    ez8_0 = cz0 * K8z;  ez8_1 = cz1 * K8z;
    exy8[0] = hx8_0 ^ hy8_0; exy8[1] = hx8_1 ^ hy8_0;
    exy8[2] = hx8_0 ^ hy8_1; exy8[3] = hx8_1 ^ hy8_1;
  } else {
    const uint32_t hy8_0 = cy0 * (res * 8u), hy8_1 = cy1 * (res * 8u);
    ez8_0 = cz0 * (res * res * 8u);  ez8_1 = cz1 * (res * res * 8u);
    exy8[0] = hx8_0 + hy8_0; exy8[1] = hx8_1 + hy8_0;
    exy8[2] = hx8_0 + hy8_1; exy8[3] = hx8_1 + hy8_1;
  }

  constexpr uint32_t mask8 = kHashMask << 3;     // 0x3FFFF8
  uint32_t boff[8];
#pragma unroll
  for (int c = 0; c < 8; ++c) {
    const uint32_t ez8 = (c & 4) ? ez8_1 : ez8_0;
    if constexpr (hashed) boff[c] = (exy8[c & 3] ^ ez8) & mask8;  // v_bitop3
    else                  boff[c] = exy8[c & 3] + ez8;
  }

  // Weights via packed f32 mults; pair halves are free to extract.
  const float wx0 = 1.0f - tx, wy0 = 1.0f - ty, wz0 = 1.0f - tz;
  const v2f wxp   = {wx0, tx};
  const v2f wxy01 = wxp * wy0;                   // {wxy0, wxy1}
  const v2f wxy23 = wxp * ty;                    // {wxy2, wxy3}
  const v2f wzp   = {wz0, tz};
  const v2f wp0 = wxy01.x * wzp;                 // {w0, w4}
  const v2f wp1 = wxy01.y * wzp;                 // {w1, w5}
  const v2f wp2 = wxy23.x * wzp;                 // {w2, w6}
  const v2f wp3 = wxy23.y * wzp;                 // {w3, w7}
  const float w[8] = {wp0.x, wp1.x, wp2.x, wp3.x, wp0.y, wp1.y, wp2.y, wp3.y};

  // Issue all 8 gathers back-to-back for memory-level parallelism, then reduce.
  v2f e[8];
#pragma unroll
  for (int c = 0; c < 8; ++c) {
    if constexpr (useLds) {
      e[c] = *(const v2f*)(ltab_bytes + (ldsByteOff + boff[c]));
    } else {
      const char* lp = gtab_bytes + (size_t)kOff[LV] * 8u;   // uniform (SGPR) base
      e[c] = *(const v2f*)(lp + (size_t)boff[c]);            // base + zext(u32)
    }
  }
  v2f acc = {0.0f, 0.0f};
#pragma unroll
  for (int c = 0; c < 8; ++c)      // corner order 0..7; 2-wide fma -> v_pk_fma_f32
    acc += w[c] * e[c];
  outv[2 * LV + 0] = acc.x;
  outv[2 * LV + 1] = acc.y;
}

__global__ __launch_bounds__(NTHREADS)
void hashenc_fwd(const float* __restrict__ in, const float* __restrict__ embf,
                 float* __restrict__ out, ScaleParams sp) {
  __shared__ alignas(16) float2 s_tab[kLdsElems];
  const char* __restrict__ gtab_bytes = reinterpret_cast<const char*>(embf);

  // ---- Stage levels 0..1 (130,112 B) into LDS via CDNA5 async global->LDS ----
#if __has_builtin(__builtin_amdgcn_global_load_async_to_lds_b128)
  {
    const uint32_t  nchunks = kLdsElems / 2u;           // 16 B per async chunk
    const uintptr_t gbase   = (uintptr_t)embf;
    const uint32_t  lbase   = (uint32_t)(uintptr_t)&s_tab[0];
    for (uint32_t c = threadIdx.x; c < nchunks; c += NTHREADS) {
      __builtin_amdgcn_global_load_async_to_lds_b128(
          (g_v4i_p)(gbase + 16u * c),
          (l_v4i_p)(uintptr_t)(lbase + 16u * c),
          0, 0);
    }
  }
 #if __has_builtin(__builtin_amdgcn_s_wait_asynccnt)
  __builtin_amdgcn_s_wait_asynccnt(0);
 #else
  asm volatile("s_wait_asynccnt 0" ::: "memory");
 #endif
#else
  {
    const float2* gt2 = reinterpret_cast<const float2*>(embf);
    for (uint32_t i = threadIdx.x; i < kLdsElems; i += NTHREADS)
      s_tab[i] = gt2[i];
  }
#endif
  __syncthreads();

  const char* lt = (const char*)&s_tab[0];
  const uint32_t base = blockIdx.x * (NTHREADS * PPT) + threadIdx.x;

#pragma unroll 1
  for (int k = 0; k < PPT; ++k) {
    const uint32_t b = base + (uint32_t)k * NTHREADS;

    if (k + 1 < PPT) __builtin_prefetch(in + 3u * (b + NTHREADS), 0, 0);

    const float x = in[3u * b + 0];
    const float y = in[3u * b + 1];
    const float z = in[3u * b + 2];

    float outv[32];
#define DO_LEVEL(LV) do_level<LV>(x, y, z, sp.s[LV], gtab_bytes, lt, outv)
    DO_LEVEL(0);  DO_LEVEL(1);  DO_LEVEL(2);  DO_LEVEL(3);
    DO_LEVEL(4);  DO_LEVEL(5);  DO_LEVEL(6);  DO_LEVEL(7);
    DO_LEVEL(8);  DO_LEVEL(9);  DO_LEVEL(10); DO_LEVEL(11);
    DO_LEVEL(12); DO_LEVEL(13); DO_LEVEL(14); DO_LEVEL(15);
#undef DO_LEVEL

    // Streamed output: 268 MB > L2, keep it non-temporal. 8 x b128 per point.
    float* op = out + (size_t)b * 32u;
#pragma unroll
    for (int q = 0; q < 8; ++q) {
      v4f v = {outv[4 * q + 0], outv[4 * q + 1], outv[4 * q + 2], outv[4 * q + 3]};
      __builtin_nontemporal_store(v, (v4f*)op + q);
    }
  }
}

extern "C" void kernel_launch(void* const* d_in, const int* in_sizes, int n_in,
                              void* d_out, int out_size, void* d_ws, size_t ws_size,
                              hipStream_t stream) {
  (void)in_sizes; (void)n_in; (void)out_size; (void)d_ws; (void)ws_size;
  const float* in  = (const float*)d_in[0];   // [B,3] f32 coords
  const float* emb = (const float*)d_in[1];   // [6098120,2] f32 embeddings
  float* out = (float*)d_out;                 // [B,32] f32

  // Exact same double-precision scale math as the reference.
  ScaleParams sp;
  const double pls_log2 = std::log2(2048.0 / 16.0) / 15.0;
  for (int l = 0; l < 16; ++l)
    sp.s[l] = (float)(std::exp2((double)l * pls_log2) * 16.0 - 1.0);

  const dim3 grid(kB / (NTHREADS * PPT));   // 512 blocks, exact cover
  hashenc_fwd<<<grid, dim3(NTHREADS), 0, stream>>>(in, emb, out, sp);
}